// Linear_40991167873156
// MI455X (gfx1250) — compile-verified
//
#include <hip/hip_runtime.h>
#include <hip/hip_bf16.h>

// Problem constants (from reference)
#define E_     4
#define RE_    64      // per-expert rank
#define DIN_   4096
#define DOUT_  4096
#define BS_    8192    // BATCH * SEQ

typedef __attribute__((ext_vector_type(2))) float v2f;
typedef __attribute__((ext_vector_type(8))) float v8f;

// ---------------------------------------------------------------------------
// Phase 1:  h[e][m][r] = sum_k x[m][k] * A[e][r][k]
//   GEMM per expert: M=8192, N=64, K=4096, fp32 via V_WMMA_F32_16X16X4_F32.
//   grid.x = BS/16 row-blocks, blockDim = 128 (4 waves; wave id == expert).
//   Each wave owns a 16x64 tile of h (4 accumulator fragments).
// ---------------------------------------------------------------------------
__global__ void __launch_bounds__(128)
lora_a_wmma(const float* __restrict__ x, const float* __restrict__ A,
            float* __restrict__ h)
{
    const int lane = threadIdx.x & 31;
    const int e    = threadIdx.x >> 5;        // expert = wave-in-block
    const int mb   = blockIdx.x;              // 16-row block of (b*s)
    const int lo   = lane & 15;
    const int hi   = lane >> 4;

    // A-fragment (16x4, MxK): lane -> row m = lo; VGPR pair -> K = kb + 2*hi {+0,+1}
    const float* xrow = x + (size_t)(mb * 16 + lo) * DIN_ + 2 * hi;
    // B-fragment (4x16, KxN): lane -> col n (= LoRA rank r); element = A_e[r][k]
    const float* arow = A + (size_t)e * RE_ * DIN_ + (size_t)lo * DIN_ + 2 * hi;

    v8f a0 = {}, a1 = {}, a2 = {}, a3 = {};

#pragma unroll 4
    for (int kb = 0; kb < DIN_; kb += 4) {
        v2f va = *(const v2f*)(xrow + kb);
        v2f b0 = *(const v2f*)(arow + (size_t)0  * 16 * DIN_ + kb);
        v2f b1 = *(const v2f*)(arow + (size_t)1  * 16 * DIN_ + kb);
        v2f b2 = *(const v2f*)(arow + (size_t)2  * 16 * DIN_ + kb);
        v2f b3 = *(const v2f*)(arow + (size_t)3  * 16 * DIN_ + kb);
        a0 = __builtin_amdgcn_wmma_f32_16x16x4_f32(false, va, false, b0, (short)0, a0, false, false);
        a1 = __builtin_amdgcn_wmma_f32_16x16x4_f32(false, va, false, b1, (short)0, a1, false, false);
        a2 = __builtin_amdgcn_wmma_f32_16x16x4_f32(false, va, false, b2, (short)0, a2, false, false);
        a3 = __builtin_amdgcn_wmma_f32_16x16x4_f32(false, va, false, b3, (short)0, a3, false, false);
    }

    // D-fragment layout: lane -> col n = lo; VGPR i -> row (i + 8*hi)
    float* hp = h + ((size_t)e * BS_ + mb * 16) * RE_;
#pragma unroll
    for (int i = 0; i < 8; ++i) {
        const int row = i + 8 * hi;
        hp[(size_t)row * RE_ +  0 + lo] = a0[i];
        hp[(size_t)row * RE_ + 16 + lo] = a1[i];
        hp[(size_t)row * RE_ + 32 + lo] = a2[i];
        hp[(size_t)row * RE_ + 48 + lo] = a3[i];
    }
}

// ---------------------------------------------------------------------------
// Phase 2:  y[e][m][n] = sum_r h[e][m][r] * B[e][n][r]
//   GEMM per expert: M=8192, N=4096, K=64.  h (8 MB) + B (4 MB) stay in L2;
//   the 512 MB y stream is written with non-temporal stores.
//   One wave computes a 16x64 tile (4 fragments).  131072 waves total.
// ---------------------------------------------------------------------------
__global__ void __launch_bounds__(256)
lora_b_wmma(const float* __restrict__ h, const float* __restrict__ Bw,
            float* __restrict__ y)
{
    const int lane = threadIdx.x & 31;
    int wid = blockIdx.x * (blockDim.x >> 5) + (threadIdx.x >> 5);

    const int NB = DOUT_ / 64;   // 64 column groups
    const int MB = BS_ / 16;     // 512 row blocks
    const int nb = wid % NB;  wid /= NB;
    const int mb = wid % MB;  wid /= MB;
    const int e  = wid;          // 0..3

    const int lo = lane & 15;
    const int hi = lane >> 4;

    // A-fragment from h: row m = lo, K pair at kb + 2*hi
    const float* hrow = h + ((size_t)e * BS_ + mb * 16 + lo) * RE_ + 2 * hi;
    // B-fragment: col n = nb*64 + t*16 + lo; element = B_e[n][k]  (row stride RE_)
    const float* brow = Bw + (size_t)e * DOUT_ * RE_
                           + (size_t)(nb * 64 + lo) * RE_ + 2 * hi;

    v8f a0 = {}, a1 = {}, a2 = {}, a3 = {};

#pragma unroll
    for (int kb = 0; kb < RE_; kb += 4) {
        v2f va = *(const v2f*)(hrow + kb);
        v2f b0 = *(const v2f*)(brow + (size_t)0 * 16 * RE_ + kb);
        v2f b1 = *(const v2f*)(brow + (size_t)1 * 16 * RE_ + kb);
        v2f b2 = *(const v2f*)(brow + (size_t)2 * 16 * RE_ + kb);
        v2f b3 = *(const v2f*)(brow + (size_t)3 * 16 * RE_ + kb);
        a0 = __builtin_amdgcn_wmma_f32_16x16x4_f32(false, va, false, b0, (short)0, a0, false, false);
        a1 = __builtin_amdgcn_wmma_f32_16x16x4_f32(false, va, false, b1, (short)0, a1, false, false);
        a2 = __builtin_amdgcn_wmma_f32_16x16x4_f32(false, va, false, b2, (short)0, a2, false, false);
        a3 = __builtin_amdgcn_wmma_f32_16x16x4_f32(false, va, false, b3, (short)0, a3, false, false);
    }

    // Store 16x64 tile; non-temporal so the 512MB y stream bypasses L2 residency
    float* yp = y + ((size_t)e * BS_ + mb * 16) * (size_t)DOUT_ + nb * 64;
#pragma unroll
    for (int i = 0; i < 8; ++i) {
        const int row = i + 8 * hi;
        float* r = yp + (size_t)row * DOUT_;
        __builtin_nontemporal_store(a0[i], r +  0 + lo);
        __builtin_nontemporal_store(a1[i], r + 16 + lo);
        __builtin_nontemporal_store(a2[i], r + 32 + lo);
        __builtin_nontemporal_store(a3[i], r + 48 + lo);
    }
}

extern "C" void kernel_launch(void* const* d_in, const int* in_sizes, int n_in,
                              void* d_out, int out_size, void* d_ws, size_t ws_size,
                              hipStream_t stream) {
    const float* x  = (const float*)d_in[0];   // [4, 2048, 4096]
    const float* A  = (const float*)d_in[1];   // [4, 64, 4096]
    const float* Bw = (const float*)d_in[2];   // [4, 4096, 64]
    float*       y  = (float*)d_out;           // [4, 4, 2048, 4096]
    float*       h  = (float*)d_ws;            // needs E*BS*RE*4 = 8 MB scratch

    // Phase 1: 512 row-blocks x 4 waves (one wave per expert)
    lora_a_wmma<<<dim3(BS_ / 16), dim3(128), 0, stream>>>(x, A, h);

    // Phase 2: 131072 waves, 8 waves per block
    const int total_waves = (BS_ / 16) * (DOUT_ / 64) * E_;  // 131072
    lora_b_wmma<<<dim3(total_waves / 8), dim3(256), 0, stream>>>(h, Bw, y);
}